// LocalRelation_4320737099852
// MI455X (gfx1250) — compile-verified
//
#include <hip/hip_runtime.h>
#include <cstddef>

// Problem constants (from reference): q,k [B=8, C=512, H=64, W=64] fp32.
#define B_   8
#define C_   512
#define H_   64
#define W_   64
#define HW_  (H_ * W_)
#define CH_  (C_ / 2)     // channels per wave-group (reduction split)

typedef __attribute__((ext_vector_type(2))) float v2f;   // A/B operand of wmma_f32_16x16x4_f32
typedef __attribute__((ext_vector_type(8))) float v8f;   // C/D accumulator (16x16 f32)

// Block = 256 threads = 8 wave32s, mapped to one (b, h) image row.
//   wave id wid: tile = wid & 3  (16-wide w tile), g = wid >> 2 (channel half).
// Each wave runs 6 WMMA accumulation chains over its 256 channels:
//   accL[dy] = Q_tile^T x K(h+dy-1, cols w0-1..w0+14)
//   accR[dy] = Q_tile^T x K(h+dy-1, cols w0+2..w0+17)
// The two channel-halves are summed during the LDS diagonal-extraction bounce.
__global__ __launch_bounds__(256) void local_relation_wmma_kernel(
    const float* __restrict__ q, const float* __restrict__ kk,
    float* __restrict__ out)
{
  const int lane = threadIdx.x & 31;
  const int wid  = threadIdx.x >> 5;          // 0..7
  const int tile = wid & 3;                   // w tile
  const int g    = wid >> 2;                  // channel half
  const int bh   = blockIdx.x;                // 0 .. B*H-1
  const int b    = bh >> 6;                   // / H_
  const int h    = bh & (H_ - 1);
  const int w0   = tile << 4;

  const int m  = lane & 15;                   // A-row / B-col index within tile
  const int hs = lane >> 4;                   // lane-half: K pair {0,1} vs {2,3}

  // ---- uniform base pointers + fixed per-lane int32 offsets ---------------
  const size_t base_off = (size_t)b * C_ * HW_ + (size_t)g * CH_ * HW_;
  const float* qp = q  + base_off;            // uniform
  const float* kp = kk + base_off;            // uniform

  const int choff = 2 * hs * HW_;             // lane-half channel offset

  // A (q): lanes 0-15 hold K=c+0 (.x) / c+1 (.y); lanes 16-31 hold K=c+2/c+3.
  const int qoff = choff + h * W_ + w0 + m;

  // B columns: left = w0 + m - 1, right = w0 + m + 2 (boundary -> zero mask).
  const int wl = w0 + m - 1;
  const int wr = w0 + m + 2;
  const bool wlv = (wl >= 0);                 // wl <= 62 always
  const bool wrv = (wr < W_);
  const int wlc = wlv ? wl : 0;               // clamped, always-valid column
  const int wrc = wrv ? wr : 0;

  const bool hv0 = (h - 1 >= 0);
  const bool hv2 = (h + 1 < H_);
  const int r0 = (hv0 ? h - 1 : 0) * W_;      // clamped, always-valid rows
  const int r1 = h * W_;
  const int r2 = (hv2 ? h + 1 : 0) * W_;

  const int oL0 = choff + r0 + wlc, oR0 = choff + r0 + wrc;
  const int oL1 = choff + r1 + wlc, oR1 = choff + r1 + wrc;
  const int oL2 = choff + r2 + wlc, oR2 = choff + r2 + wrc;

  // Zero-pad masks as floats: branchless boundary handling (dual-issue v_mul).
  const float mL0 = (hv0 & wlv) ? 1.0f : 0.0f;
  const float mR0 = (hv0 & wrv) ? 1.0f : 0.0f;
  const float mL1 = wlv ? 1.0f : 0.0f;
  const float mR1 = wrv ? 1.0f : 0.0f;
  const float mL2 = (hv2 & wlv) ? 1.0f : 0.0f;
  const float mR2 = (hv2 & wrv) ? 1.0f : 0.0f;

  v8f aL0 = {0,0,0,0,0,0,0,0}, aR0 = {0,0,0,0,0,0,0,0};
  v8f aL1 = {0,0,0,0,0,0,0,0}, aR1 = {0,0,0,0,0,0,0,0};
  v8f aL2 = {0,0,0,0,0,0,0,0}, aR2 = {0,0,0,0,0,0,0,0};

  const size_t cstep = (size_t)4 * HW_;       // 4 channels / iteration

  // unroll 4: four chunks share one address register per stream via the
  // 24-bit immediate offset (max 0x34000 bytes << +-8MB), amortizing the
  // v_add_nc_u64 pointer advances and lengthening the load clauses.
  #pragma unroll 4
  for (int c = 0; c < CH_; c += 4) {
    // Issue all 14 loads unconditionally (clause-able, partial waits).
    const float ax  = qp[qoff];       const float ay  = qp[qoff + HW_];
    const float l0x = kp[oL0];        const float l0y = kp[oL0 + HW_];
    const float r0x = kp[oR0];        const float r0y = kp[oR0 + HW_];
    const float l1x = kp[oL1];        const float l1y = kp[oL1 + HW_];
    const float r1x = kp[oR1];        const float r1y = kp[oR1 + HW_];
    const float l2x = kp[oL2];        const float l2y = kp[oL2 + HW_];
    const float r2x = kp[oR2];        const float r2y = kp[oR2 + HW_];

    v2f a;  a.x = ax;  a.y = ay;
    v2f bv;
    bv.x = l0x * mL0;  bv.y = l0y * mL0;
    aL0 = __builtin_amdgcn_wmma_f32_16x16x4_f32(false, a, false, bv, (short)0, aL0, false, false);
    bv.x = r0x * mR0;  bv.y = r0y * mR0;
    aR0 = __builtin_amdgcn_wmma_f32_16x16x4_f32(false, a, false, bv, (short)0, aR0, false, false);
    bv.x = l1x * mL1;  bv.y = l1y * mL1;
    aL1 = __builtin_amdgcn_wmma_f32_16x16x4_f32(false, a, false, bv, (short)0, aL1, false, false);
    bv.x = r1x * mR1;  bv.y = r1y * mR1;
    aR1 = __builtin_amdgcn_wmma_f32_16x16x4_f32(false, a, false, bv, (short)0, aR1, false, false);
    bv.x = l2x * mL2;  bv.y = l2y * mL2;
    aL2 = __builtin_amdgcn_wmma_f32_16x16x4_f32(false, a, false, bv, (short)0, aL2, false, false);
    bv.x = r2x * mR2;  bv.y = r2y * mR2;
    aR2 = __builtin_amdgcn_wmma_f32_16x16x4_f32(false, a, false, bv, (short)0, aR2, false, false);

    qp += cstep;                              // uniform scalar advance
    kp += cstep;
  }

  // ---- diagonal extraction via LDS bounce (also sums the two C-halves) ----
  // C/D layout: VGPR j, lanes 0-15 -> G[M=j][N=lane]; lanes 16-31 -> G[M=8+j][N=lane-16].
  // So G[M][N] lives at lds[..][M&7][N + 16*(M>=8)].
  __shared__ float lds[2][4][3][2][8][32];    // 48 KB / block
  #pragma unroll
  for (int j = 0; j < 8; ++j) {
    lds[g][tile][0][0][j][lane] = aL0[j];
    lds[g][tile][0][1][j][lane] = aR0[j];
    lds[g][tile][1][0][j][lane] = aL1[j];
    lds[g][tile][1][1][j][lane] = aR1[j];
    lds[g][tile][2][0][j][lane] = aL2[j];
    lds[g][tile][2][1][j][lane] = aR2[j];
  }
  __syncthreads();

  if (wid < 4 && lane < 16) {                 // g==0 waves; tile == wid
    const int pos = bh * W_ + w0 + lane;      // flat (b,h,w)
    float* op = out + (size_t)pos * 9;
    const int jr = lane & 7;
    const int lh = (lane >> 3) << 4;          // +16 if m >= 8
    #pragma unroll
    for (int dy = 0; dy < 3; ++dy) {
      #pragma unroll
      for (int dx = 0; dx < 3; ++dx) {
        const int n = lane + dx;              // wanted Gram column
        float e;
        if (n < 16)
          e = lds[0][tile][dy][0][jr][n + lh] + lds[1][tile][dy][0][jr][n + lh];
        else
          e = lds[0][tile][dy][1][jr][(n - 3) + lh] + lds[1][tile][dy][1][jr][(n - 3) + lh];
        op[dy * 3 + dx] = 1.0f / (1.0f + __expf(-e));   // sigmoid
      }
    }
  }
}

extern "C" void kernel_launch(void* const* d_in, const int* in_sizes, int n_in,
                              void* d_out, int out_size, void* d_ws, size_t ws_size,
                              hipStream_t stream) {
  const float* q  = (const float*)d_in[0];
  const float* k  = (const float*)d_in[1];
  float* out      = (float*)d_out;

  dim3 grid(B_ * H_);       // 512 blocks: one per (b, h) row
  dim3 block(256);          // 8 wave32s: 4 w-tiles x 2 channel-halves
  hipLaunchKernelGGL(local_relation_wmma_kernel, grid, block, 0, stream, q, k, out);
}